// GAT_1_64252710748440
// MI455X (gfx1250) — compile-verified
//
#include <hip/hip_runtime.h>
#include <hip/hip_bf16.h>
#include <math.h>

typedef float v2f __attribute__((ext_vector_type(2)));
typedef float v8f __attribute__((ext_vector_type(8)));
typedef int v4i __attribute__((vector_size(16)));    // matches async-LDS builtin param

#define NEG_SLOPE 0.2f

__device__ __forceinline__ float lrelu(float x) { return x > 0.f ? x : NEG_SLOPE * x; }

// Order-preserving float<->uint mapping for atomicMax-based segment max.
__device__ __forceinline__ unsigned ford(float f) {
    unsigned u = __float_as_uint(f);
    return (u & 0x80000000u) ? ~u : (u | 0x80000000u);
}
__device__ __forceinline__ float funord(unsigned o) {
    unsigned u = (o & 0x80000000u) ? (o & 0x7FFFFFFFu) : ~o;
    return __uint_as_float(u);
}

#if __has_builtin(__builtin_amdgcn_global_load_async_to_lds_b128) && \
    __has_builtin(__builtin_amdgcn_s_wait_asynccnt)
#define USE_ASYNC_LDS 1
#else
#define USE_ASYNC_LDS 0
#endif

// ---------------------------------------------------------------------------
// GEMM: C[M, Ncols] = A[M, 128] @ B[128, Ncols], one wave per 16x16 tile,
// V_WMMA_F32_16X16X4_F32 (full f32 precision). B (the weight matrix) is
// staged into LDS once per block via async global->LDS DMA (ASYNCcnt path).
// ---------------------------------------------------------------------------
__global__ void gemm_wmma_f32(const float* __restrict__ A, const float* __restrict__ B,
                              float* __restrict__ C, int nNtiles, int Ncols, int totalTiles) {
    extern __shared__ float Bs[];                    // 128 * Ncols floats
    const int nElems = 128 * Ncols;

    // Cooperative stage of B into LDS (16B per lane per op, ASYNCcnt-tracked).
#if USE_ASYNC_LDS
    for (int i = threadIdx.x * 4; i < nElems; i += blockDim.x * 4) {
        __builtin_amdgcn_global_load_async_to_lds_b128(
            (__attribute__((address_space(1))) v4i*)(B + i),
            (__attribute__((address_space(3))) v4i*)(Bs + i),
            /*offset=*/0, /*cpol=*/0);
    }
    __builtin_amdgcn_s_wait_asynccnt(0);
#else
    for (int i = threadIdx.x * 4; i < nElems; i += blockDim.x * 4) {
        *(float4*)(Bs + i) = *(const float4*)(B + i);
    }
#endif
    __syncthreads();

    int wave = blockIdx.x * (blockDim.x >> 5) + (threadIdx.x >> 5);
    if (wave < totalTiles) {                         // wave-uniform: EXEC all-ones inside
        int mt = wave / nNtiles, nt = wave % nNtiles;
        int row0 = mt << 4, col0 = nt << 4;

        int lane = threadIdx.x & 31;
        int half = lane >> 4;                        // K sub-block select
        int l    = lane & 15;
        int col  = col0 + l;
        bool cok = col < Ncols;

        const float* arow = A + (size_t)(row0 + l) * 128 + 2 * half;
        const float* bcol = Bs + (size_t)(2 * half) * Ncols + (cok ? col : 0);

        v8f acc = {};
        #pragma unroll 4
        for (int k0 = 0; k0 < 128; k0 += 4) {
            __builtin_prefetch(arow + 16, 0, 0);     // global_prefetch_b8 one line ahead
            v2f a; a.x = arow[0]; a.y = arow[1];     // A[l, k0+2h], A[l, k0+2h+1]
            float b0 = bcol[0], b1 = bcol[Ncols];    // Bs[k0+2h, col], Bs[k0+2h+1, col]
            v2f b; b.x = cok ? b0 : 0.f; b.y = cok ? b1 : 0.f;
            acc = __builtin_amdgcn_wmma_f32_16x16x4_f32(
                /*neg_a=*/false, a, /*neg_b=*/false, b,
                /*c_mod=*/(short)0, acc, /*reuse_a=*/false, /*reuse_b=*/false);
            arow += 4;
            bcol += (size_t)4 * Ncols;
        }
        if (cok) {
            float* crow = C + (size_t)(row0 + half * 8) * Ncols + col;
            #pragma unroll
            for (int r = 0; r < 8; ++r) crow[(size_t)r * Ncols] = acc[r];
        }
    }
}

// ---------------------------------------------------------------------------
// Attention pre-pass: per (node, head) a_src/a_dst dots + self-loop max init.
// ---------------------------------------------------------------------------
__global__ void attn_pre(const float* __restrict__ H, const float* __restrict__ asv,
                         const float* __restrict__ adv, float* __restrict__ as_,
                         float* __restrict__ ad_, unsigned* __restrict__ mord,
                         int N, int heads, int ch) {
    int idx = blockIdx.x * blockDim.x + threadIdx.x;
    if (idx >= N * heads) return;
    int n = idx / heads, h = idx % heads;
    const float* hr = H + (size_t)n * heads * ch + (size_t)h * ch;
    float s = 0.f, d = 0.f;
    for (int c = 0; c < ch; ++c) {
        float v = hr[c];
        s = fmaf(v, asv[h * ch + c], s);
        d = fmaf(v, adv[h * ch + c], d);
    }
    as_[idx] = s;
    ad_[idx] = d;
    mord[idx] = ford(lrelu(s + d));                  // self-loop seeds segment max
}

__global__ void edge_max(const int* __restrict__ ei, const float* __restrict__ as_,
                         const float* __restrict__ ad_, unsigned* __restrict__ mord,
                         int E, int heads) {
    int idx = blockIdx.x * blockDim.x + threadIdx.x;
    if (idx >= E * heads) return;
    int eid = idx / heads, h = idx % heads;
    int src = ei[eid], dst = ei[E + eid];
    float e = lrelu(as_[src * heads + h] + ad_[dst * heads + h]);
    atomicMax(&mord[dst * heads + h], ford(e));
}

// Decode max in place to float; seed sum with self-loop exp term.
__global__ void init_sum(unsigned* __restrict__ mord, const float* __restrict__ as_,
                         const float* __restrict__ ad_, float* __restrict__ s, int NH) {
    int idx = blockIdx.x * blockDim.x + threadIdx.x;
    if (idx >= NH) return;
    float mf = funord(mord[idx]);
    float es = lrelu(as_[idx] + ad_[idx]);
    ((float*)mord)[idx] = mf;
    s[idx] = expf(es - mf);
}

__global__ void edge_sum(const int* __restrict__ ei, const float* __restrict__ as_,
                         const float* __restrict__ ad_, const float* __restrict__ mf,
                         float* __restrict__ s, int E, int heads) {
    int idx = blockIdx.x * blockDim.x + threadIdx.x;
    if (idx >= E * heads) return;
    int eid = idx / heads, h = idx % heads;
    int src = ei[eid], dst = ei[E + eid];
    float e = lrelu(as_[src * heads + h] + ad_[dst * heads + h]);
    atomicAdd(&s[dst * heads + h], expf(e - mf[dst * heads + h]));
}

// Seed aggregation with self-loop contribution alpha_self * H[n].
__global__ void init_agg(const float* __restrict__ H, const float* __restrict__ as_,
                         const float* __restrict__ ad_, const float* __restrict__ mf,
                         const float* __restrict__ s, float* __restrict__ AGG,
                         int N, int heads, int ch) {
    int idx = blockIdx.x * blockDim.x + threadIdx.x;
    int hc = heads * ch;
    if (idx >= N * hc) return;
    int n = idx / hc, c = idx % hc, h = c / ch;
    float es = lrelu(as_[n * heads + h] + ad_[n * heads + h]);
    float alpha = expf(es - mf[n * heads + h]) / (s[n * heads + h] + 1e-16f);
    AGG[idx] = alpha * H[idx];
}

// Layer-1 aggregation: one wave per edge. Lane l owns channels 4l..4l+3, so
// its head h = l/8 is unique per lane -> exactly one expf per lane.
// Edge indices are wave-uniform -> scalarized loads.
__global__ void edge_agg128(const int* __restrict__ ei, const float* __restrict__ H,
                            const float* __restrict__ as_, const float* __restrict__ ad_,
                            const float* __restrict__ mf, const float* __restrict__ s,
                            float* __restrict__ AGG, int E) {
    int eid = blockIdx.x * (blockDim.x >> 5) + (threadIdx.x >> 5);
    if (eid >= E) return;
    int lane = threadIdx.x & 31;
    int src = ei[eid], dst = ei[E + eid];
    int h = lane >> 3;
    float e = lrelu(as_[src * 4 + h] + ad_[dst * 4 + h]);
    float alpha = expf(e - mf[dst * 4 + h]) / (s[dst * 4 + h] + 1e-16f);
    float4 hv = *(const float4*)(H + (size_t)src * 128 + lane * 4);
    float* dp = AGG + (size_t)dst * 128 + lane * 4;
    atomicAdd(dp + 0, alpha * hv.x);
    atomicAdd(dp + 1, alpha * hv.y);
    atomicAdd(dp + 2, alpha * hv.z);
    atomicAdd(dp + 3, alpha * hv.w);
}

// Layer-2 aggregation: one wave per edge, heads=1, 40 channels.
__global__ void edge_agg40(const int* __restrict__ ei, const float* __restrict__ H,
                           const float* __restrict__ as_, const float* __restrict__ ad_,
                           const float* __restrict__ mf, const float* __restrict__ s,
                           float* __restrict__ AGG, int E) {
    int eid = blockIdx.x * (blockDim.x >> 5) + (threadIdx.x >> 5);
    if (eid >= E) return;
    int lane = threadIdx.x & 31;
    int src = ei[eid], dst = ei[E + eid];
    float e = lrelu(as_[src] + ad_[dst]);
    float alpha = expf(e - mf[dst]) / (s[dst] + 1e-16f);
    const float* hs = H + (size_t)src * 40;
    float* dp = AGG + (size_t)dst * 40;
    for (int c = lane; c < 40; c += 32)
        atomicAdd(dp + c, alpha * hs[c]);
}

// Layer-1 epilogue: bias + ELU (in place).
__global__ void bias_elu(float* __restrict__ A, const float* __restrict__ b, int N, int D) {
    int idx = blockIdx.x * blockDim.x + threadIdx.x;
    if (idx >= N * D) return;
    float v = A[idx] + b[idx % D];
    A[idx] = v > 0.f ? v : expm1f(v);
}

// Layer-2 epilogue: bias + row-wise log_softmax into d_out.
__global__ void bias_logsoftmax(const float* __restrict__ A, const float* __restrict__ b,
                                float* __restrict__ out, int N, int D) {
    int n = blockIdx.x * blockDim.x + threadIdx.x;
    if (n >= N) return;
    const float* row = A + (size_t)n * D;
    float mx = -INFINITY;
    for (int c = 0; c < D; ++c) mx = fmaxf(mx, row[c] + b[c]);
    float sum = 0.f;
    for (int c = 0; c < D; ++c) sum += expf(row[c] + b[c] - mx);
    float lse = mx + logf(sum);
    float* orow = out + (size_t)n * D;
    for (int c = 0; c < D; ++c) orow[c] = row[c] + b[c] - lse;
}

// ---------------------------------------------------------------------------
extern "C" void kernel_launch(void* const* d_in, const int* in_sizes, int n_in,
                              void* d_out, int out_size, void* d_ws, size_t ws_size,
                              hipStream_t stream) {
    const float* x   = (const float*)d_in[0];
    const int*   ei  = (const int*)d_in[1];
    const float* W1  = (const float*)d_in[2];
    const float* aS1 = (const float*)d_in[3];
    const float* aD1 = (const float*)d_in[4];
    const float* b1  = (const float*)d_in[5];
    const float* W2  = (const float*)d_in[6];
    const float* aS2 = (const float*)d_in[7];
    const float* aD2 = (const float*)d_in[8];
    const float* b2  = (const float*)d_in[9];
    float* out = (float*)d_out;

    const int IN = 128, HID = 32, H = 4, OUT = 40, D1 = H * HID; // 128
    const int N = in_sizes[0] / IN;
    const int E = in_sizes[1] / 2;

    // Workspace layout (floats).
    float* ws   = (float*)d_ws;
    float* H1   = ws;                                // N*128
    float* AGG1 = ws + (size_t)N * D1;               // N*128
    float* as1  = AGG1 + (size_t)N * D1;             // N*4
    float* ad1  = as1 + (size_t)N * H;
    unsigned* m1 = (unsigned*)(ad1 + (size_t)N * H); // ordered-uint, then float
    float* s1   = (float*)m1 + (size_t)N * H;
    // Layer 2 reuses the H1 region once layer 1 is done with it.
    float* G    = H1;                                // N*40
    float* AGG2 = H1 + (size_t)N * OUT;              // N*40
    float* as2  = AGG2 + (size_t)N * OUT;            // N
    float* ad2  = as2 + N;
    unsigned* m2 = (unsigned*)(ad2 + N);
    float* s2   = (float*)m2 + N;

    const int B = 256;
    auto g = [](long long work, int blk) { return (unsigned)((work + blk - 1) / blk); };

    // ---- Layer 1 ----
    int tiles1 = (N / 16) * (D1 / 16);               // 6250 * 8
    gemm_wmma_f32<<<g((long long)tiles1 * 32, B), B, 128 * D1 * sizeof(float), stream>>>(
        x, W1, H1, D1 / 16, D1, tiles1);

    attn_pre<<<g((long long)N * H, B), B, 0, stream>>>(H1, aS1, aD1, as1, ad1, m1, N, H, HID);
    edge_max<<<g((long long)E * H, B), B, 0, stream>>>(ei, as1, ad1, m1, E, H);
    init_sum<<<g((long long)N * H, B), B, 0, stream>>>(m1, as1, ad1, s1, N * H);
    edge_sum<<<g((long long)E * H, B), B, 0, stream>>>(ei, as1, ad1, (float*)m1, s1, E, H);
    init_agg<<<g((long long)N * D1, B), B, 0, stream>>>(H1, as1, ad1, (float*)m1, s1, AGG1, N, H, HID);
    edge_agg128<<<g((long long)E * 32, B), B, 0, stream>>>(ei, H1, as1, ad1, (float*)m1, s1, AGG1, E);
    bias_elu<<<g((long long)N * D1, B), B, 0, stream>>>(AGG1, b1, N, D1);

    // ---- Layer 2 (heads=1, ch=40) ----
    int nNt2 = (OUT + 15) / 16;                      // 3 column tiles (48, masked to 40)
    int tiles2 = (N / 16) * nNt2;
    gemm_wmma_f32<<<g((long long)tiles2 * 32, B), B, 128 * OUT * sizeof(float), stream>>>(
        AGG1, W2, G, nNt2, OUT, tiles2);

    attn_pre<<<g((long long)N, B), B, 0, stream>>>(G, aS2, aD2, as2, ad2, m2, N, 1, OUT);
    edge_max<<<g((long long)E, B), B, 0, stream>>>(ei, as2, ad2, m2, E, 1);
    init_sum<<<g((long long)N, B), B, 0, stream>>>(m2, as2, ad2, s2, N);
    edge_sum<<<g((long long)E, B), B, 0, stream>>>(ei, as2, ad2, (float*)m2, s2, E, 1);
    init_agg<<<g((long long)N * OUT, B), B, 0, stream>>>(G, as2, ad2, (float*)m2, s2, AGG2, N, 1, OUT);
    edge_agg40<<<g((long long)E * 32, B), B, 0, stream>>>(ei, G, as2, ad2, (float*)m2, s2, AGG2, E);
    bias_logsoftmax<<<g((long long)N, B), B, 0, stream>>>(AGG2, b2, out, N, OUT);
}